// StochasticDurationPredictor_4939212390579
// MI455X (gfx1250) — compile-verified
//
#include <hip/hip_runtime.h>
#include <math.h>

// ---------------------------------------------------------------------------
// MI455X (gfx1250) implementation of the StochasticDurationPredictor forward.
// Dominant cost: 33x fp32 GEMM (192x192 x 65536) -> V_WMMA_F32_16X16X4_F32.
// Activations (50MB) are L2-resident (192MB L2); elementwise ops fused.
// GEMM tiling: each wave owns one 16-wide N tile and MTILES/2 M tiles, so the
// B fragment (LDS) is loaded once per K-step and reused across 6 WMMAs.
// LDS tiles are filled with GLOBAL_LOAD_ASYNC_TO_LDS_B128 (ASYNCcnt path).
// ---------------------------------------------------------------------------

#define CC    192
#define TT    1024
#define L2PI  1.8378770664093454f
#define INV_SQRT_C 0.07216878364870323f   // 1/sqrt(192)

typedef __attribute__((ext_vector_type(2))) float v2f;
typedef __attribute__((ext_vector_type(8))) float v8f;

// direct global->LDS async path if the toolchain exposes it (gfx1250)
#if defined(__HIP_DEVICE_COMPILE__) && __has_builtin(__builtin_amdgcn_global_load_async_to_lds_b128) && __has_builtin(__builtin_amdgcn_s_wait_asynccnt)
#define ASYNC_LDS 1
#else
#define ASYNC_LDS 0
#endif

#if ASYNC_LDS
// builtin expects: (int4-gcc-vector AS1*, ... AS3*, imm offset, imm cpol)
typedef int async_i4 __attribute__((vector_size(4 * sizeof(int))));
__device__ __forceinline__ void asyncCopy16(const float* g, float* l) {
    async_i4* gv = (async_i4*)g;                 // reinterpret in generic AS
    async_i4* lv = (async_i4*)l;
    __builtin_amdgcn_global_load_async_to_lds_b128(
        (__attribute__((address_space(1))) async_i4*)gv,   // flat -> global
        (__attribute__((address_space(3))) async_i4*)lv,   // flat -> LDS
        0, 0);
}
#endif

// ------------------------------ device helpers -----------------------------

__device__ __forceinline__ float geluf(float x) {        // exact gelu
    return 0.5f * x * (1.0f + erff(x * 0.70710678118654752f));
}
__device__ __forceinline__ float softplusf(float x) {    // stable softplus
    return fmaxf(x, 0.0f) + log1pf(expf(-fabsf(x)));
}
__device__ __forceinline__ float logsigf(float x) { return -softplusf(-x); }
__device__ __forceinline__ float sigf(float x)    { return 1.0f / (1.0f + expf(-x)); }

__device__ __forceinline__ float waveReduce(float v) {   // wave32 reduce
#pragma unroll
    for (int off = 16; off > 0; off >>= 1) v += __shfl_down(v, off, 32);
    return v;
}
// 256-thread block sum -> one global f32 atomic (global_atomic_add_f32)
__device__ __forceinline__ void blockAtomicAdd(float* dst, float v, float* sbuf) {
    const int tid = threadIdx.x;
    v = waveReduce(v);
    if ((tid & 31) == 0) sbuf[tid >> 5] = v;
    __syncthreads();
    if (tid == 0) {
        float s = 0.f;
#pragma unroll
        for (int i = 0; i < 8; ++i) s += sbuf[i];
        atomicAdd(dst, s);
    }
    __syncthreads();
}

// ------------------------- WMMA fp32 1x1-conv GEMM -------------------------
// Y[b,o,t] = sum_c W[o,c] * X[b,c,t] + bias[o]   (optionally * mask[b,t])
// Block = 256 threads (8 waves). Grid = (TT/64, B). K = CC = 192, step 4.
// wave -> (nt = wave&3, mhalf = wave>>2); each wave holds MTILES/2 accums.
template <int MTILES, int CT, bool MASK>
__global__ __launch_bounds__(256) void k_conv1x1_wmma(
    const float* __restrict__ X, const float* __restrict__ W,
    const float* __restrict__ bias, const float* __restrict__ mask,
    float* __restrict__ Y)
{
    constexpr int LDT  = 68;                 // padded LDS stride (floats)
    constexpr int MTW  = MTILES / 2;         // M tiles per wave
    constexpr bool FULL = (CT == MTILES * 16);
    __shared__ float sX[CC * LDT];           // 52,224 bytes
    const int b    = blockIdx.y;
    const int t0   = blockIdx.x * 64;
    const int tid  = threadIdx.x;
    const int lane = tid & 31;
    const int wave = tid >> 5;

    // stage X[b, 0:192, t0:t0+64] into LDS
    const float* Xb = X + (size_t)b * CC * TT + t0;
#if ASYNC_LDS
    for (int i = tid; i < CC * 16; i += 256) {
        const int c  = i >> 4;
        const int tq = (i & 15) << 2;
        asyncCopy16(Xb + (size_t)c * TT + tq, &sX[c * LDT + tq]);
    }
    __builtin_amdgcn_s_wait_asynccnt(0);
    __syncthreads();
#else
    for (int i = tid; i < CC * 16; i += 256) {
        const int c  = i >> 4;
        const int tq = (i & 15) << 2;
        const float4 v = *reinterpret_cast<const float4*>(Xb + (size_t)c * TT + tq);
        float* d = &sX[c * LDT + tq];
        d[0] = v.x; d[1] = v.y; d[2] = v.z; d[3] = v.w;
    }
    __syncthreads();
#endif

    const int hi   = lane >> 4;              // half-wave select (ISA K split)
    const int l16  = lane & 15;
    const int nt   = wave & 3;
    const int mh   = wave >> 2;              // 0 or 1
    const int tcol = nt * 16 + l16;          // local t column

#pragma unroll
    for (int mi = 0; mi < MTW; ++mi)
        __builtin_prefetch(W + (size_t)((mh * MTW + mi) * 16) * CC, 0, 0);

    v8f acc[MTW];
#pragma unroll
    for (int mi = 0; mi < MTW; ++mi) acc[mi] = (v8f){};

    for (int k = 0; k < CC; k += 4) {
        const int kk = k + hi * 2;
        v2f bf;                               // B: (K=kk/kk+1, N=tcol)
        bf.x = sX[(kk)     * LDT + tcol];
        bf.y = sX[(kk + 1) * LDT + tcol];
#pragma unroll
        for (int mi = 0; mi < MTW; ++mi) {
            const int arow = (mh * MTW + mi) * 16 + l16;   // output row (o)
            v2f a;
            if (FULL) {
                const float2 wv = *reinterpret_cast<const float2*>(W + (size_t)arow * CC + kk);
                a.x = wv.x; a.y = wv.y;
            } else {
                const bool rowok = (arow < CT);
                const float2 wv = *reinterpret_cast<const float2*>(
                    W + (size_t)(rowok ? arow : 0) * CC + kk);
                a.x = rowok ? wv.x : 0.f;
                a.y = rowok ? wv.y : 0.f;
            }
            acc[mi] = __builtin_amdgcn_wmma_f32_16x16x4_f32(
                false, a, false, bf, (short)0, acc[mi], false, false);
        }
    }

    const float mval = MASK ? mask[(size_t)b * TT + t0 + tcol] : 1.f;
#pragma unroll
    for (int mi = 0; mi < MTW; ++mi) {
#pragma unroll
        for (int r = 0; r < 8; ++r) {
            const int row = (mh * MTW + mi) * 16 + r + hi * 8;  // D VGPR map
            if (FULL || row < CT) {
                float v = acc[mi][r] + bias[row];
                if (MASK) v *= mval;
                Y[((size_t)b * CT + row) * TT + t0 + tcol] = v;
            }
        }
    }
}

// ---------------- fused: mask * depthwise conv -> LN1 -> GELU --------------
// Block = (b, 64 t's). Grid = (TT/64, B).
__global__ __launch_bounds__(256) void k_dw_ln_gelu(
    const float* __restrict__ X, const float* __restrict__ mask,
    const float* __restrict__ dw_w, const float* __restrict__ dw_b,
    const float* __restrict__ g1, const float* __restrict__ b1,
    float* __restrict__ Y, int dil)
{
    __shared__ float sY[CC * 64];           // 48KB
    __shared__ float spart[2][4][64];
    __shared__ float sstat[2][64];
    const int b   = blockIdx.y;
    const int t0  = blockIdx.x * 64;
    const int tid = threadIdx.x;

    for (int i = tid; i < CC * 64; i += 256) {
        const int c  = i >> 6;
        const int tl = i & 63;
        const int t  = t0 + tl;
        float acc = dw_b[c];
#pragma unroll
        for (int j = 0; j < 3; ++j) {
            const int tt = t + (j - 1) * dil;
            if (tt >= 0 && tt < TT)
                acc += dw_w[c * 3 + j] *
                       X[((size_t)b * CC + c) * TT + tt] * mask[(size_t)b * TT + tt];
        }
        sY[c * 64 + tl] = acc;
    }
    __syncthreads();
    {   // layernorm over C per column
        const int part = tid >> 6, col = tid & 63;
        float s = 0.f, s2 = 0.f;
        for (int c = part * 48; c < part * 48 + 48; ++c) {
            const float v = sY[c * 64 + col];
            s += v; s2 += v * v;
        }
        spart[0][part][col] = s; spart[1][part][col] = s2;
    }
    __syncthreads();
    if (tid < 64) {
        float s = 0.f, s2 = 0.f;
#pragma unroll
        for (int p = 0; p < 4; ++p) { s += spart[0][p][tid]; s2 += spart[1][p][tid]; }
        const float mean = s * (1.0f / CC);
        const float var  = s2 * (1.0f / CC) - mean * mean;
        sstat[0][tid] = mean;
        sstat[1][tid] = rsqrtf(var + 1e-5f);
    }
    __syncthreads();
    for (int i = tid; i < CC * 64; i += 256) {
        const int c  = i >> 6;
        const int tl = i & 63;
        const float v = (sY[c * 64 + tl] - sstat[0][tl]) * sstat[1][tl] * g1[c] + b1[c];
        Y[((size_t)b * CC + c) * TT + t0 + tl] = geluf(v);
    }
}

// ---------------- fused: LN2 -> GELU -> residual add (+ final mask) --------
__global__ __launch_bounds__(256) void k_ln_gelu_add(
    const float* __restrict__ P, const float* __restrict__ g2,
    const float* __restrict__ b2, const float* __restrict__ mask,
    float* __restrict__ X, int finalMask)
{
    __shared__ float sP[CC * 64];
    __shared__ float spart[2][4][64];
    __shared__ float sstat[2][64];
    const int b   = blockIdx.y;
    const int t0  = blockIdx.x * 64;
    const int tid = threadIdx.x;

    const float* Pb = P + (size_t)b * CC * TT + t0;
#if ASYNC_LDS
    for (int i = tid; i < CC * 16; i += 256) {
        const int c  = i >> 4;
        const int tq = (i & 15) << 2;
        asyncCopy16(Pb + (size_t)c * TT + tq, &sP[c * 64 + tq]);
    }
    __builtin_amdgcn_s_wait_asynccnt(0);
    __syncthreads();
#else
    for (int i = tid; i < CC * 16; i += 256) {
        const int c  = i >> 4;
        const int tq = (i & 15) << 2;
        const float4 v = *reinterpret_cast<const float4*>(Pb + (size_t)c * TT + tq);
        float* d = &sP[c * 64 + tq];
        d[0] = v.x; d[1] = v.y; d[2] = v.z; d[3] = v.w;
    }
    __syncthreads();
#endif
    {
        const int part = tid >> 6, col = tid & 63;
        float s = 0.f, s2 = 0.f;
        for (int c = part * 48; c < part * 48 + 48; ++c) {
            const float v = sP[c * 64 + col];
            s += v; s2 += v * v;
        }
        spart[0][part][col] = s; spart[1][part][col] = s2;
    }
    __syncthreads();
    if (tid < 64) {
        float s = 0.f, s2 = 0.f;
#pragma unroll
        for (int p = 0; p < 4; ++p) { s += spart[0][p][tid]; s2 += spart[1][p][tid]; }
        const float mean = s * (1.0f / CC);
        const float var  = s2 * (1.0f / CC) - mean * mean;
        sstat[0][tid] = mean;
        sstat[1][tid] = rsqrtf(var + 1e-5f);
    }
    __syncthreads();
    for (int i = tid; i < CC * 64; i += 256) {
        const int c  = i >> 6;
        const int tl = i & 63;
        const float v = geluf((sP[c * 64 + tl] - sstat[0][tl]) * sstat[1][tl] * g2[c] + b2[c]);
        const size_t gi = ((size_t)b * CC + c) * TT + t0 + tl;
        float r = X[gi] + v;
        if (finalMask) r *= mask[(size_t)b * TT + t0 + tl];
        X[gi] = r;
    }
}

// ----------------- rank-1 expand: Y[b,c,t] = w1[c]*S[b,t]+b1[c] (+G) -------
__global__ __launch_bounds__(256) void k_expand_add(
    const float* __restrict__ S, int sStrideB,
    const float* __restrict__ w1, const float* __restrict__ b1,
    const float* __restrict__ G, float* __restrict__ Y, long long N)
{
    const long long idx = (long long)blockIdx.x * 256 + threadIdx.x;
    if (idx >= N) return;
    const int t = (int)(idx % TT);
    const int c = (int)((idx / TT) % CC);
    const int b = (int)(idx / ((long long)CC * TT));
    float v = w1[c] * S[(size_t)b * sStrideB + t] + b1[c];
    if (G) v += G[idx];
    Y[idx] = v;
}

__global__ __launch_bounds__(256) void k_add2(
    const float* __restrict__ A, const float* __restrict__ B2,
    float* __restrict__ Y, long long N)
{
    const long long idx = (long long)blockIdx.x * 256 + threadIdx.x;
    if (idx < N) Y[idx] = A[idx] + B2[idx];
}

__global__ __launch_bounds__(256) void k_zero(float* __restrict__ a, int n)
{
    for (int i = threadIdx.x; i < n; i += 256) a[i] = 0.f;
}

// ----------------- z_q init + logdet_q / logq gaussian terms ---------------
__global__ __launch_bounds__(256) void k_zq_init(
    const float* __restrict__ EQ, const float* __restrict__ pm,
    const float* __restrict__ pl, const float* __restrict__ mask,
    float* __restrict__ Z, float* __restrict__ accLdq, float* __restrict__ accGq)
{
    __shared__ float sred[8];
    const int b = blockIdx.y;
    const int t = blockIdx.x * 256 + threadIdx.x;
    const float m  = mask[(size_t)b * TT + t];
    const float e0 = EQ[((size_t)b * 2 + 0) * TT + t];
    const float e1 = EQ[((size_t)b * 2 + 1) * TT + t];
    const float pl0 = pl[0], pl1 = pl[1];
    Z[((size_t)b * 2 + 0) * TT + t] = (pm[0] + expf(pl0) * e0) * m;
    Z[((size_t)b * 2 + 1) * TT + t] = (pm[1] + expf(pl1) * e1) * m;
    blockAtomicAdd(&accLdq[b], (pl0 + pl1) * m, sred);
    blockAtomicAdd(&accGq[b], -0.5f * (2.f * L2PI + e0 * e0 + e1 * e1) * m, sred);
}

// --------- rational-quadratic spline on zb, in-place channel flip ----------
__global__ __launch_bounds__(256) void k_spline(
    float* __restrict__ Z, const float* __restrict__ H,
    const float* __restrict__ mask, float* __restrict__ acc)
{
    __shared__ float sred[8];
    const int b = blockIdx.y;
    const int t = blockIdx.x * 256 + threadIdx.x;
    const float m  = mask[(size_t)b * TT + t];
    const float za = Z[((size_t)b * 2 + 0) * TT + t];
    const float zb = Z[((size_t)b * 2 + 1) * TT + t];

    float hj[29];
#pragma unroll
    for (int j = 0; j < 29; ++j) hj[j] = H[((size_t)b * 29 + j) * TT + t];

    float cw[11], ch[11], dv[11];
    {   // widths: softmax(hj[0:10]/sqrt(C)) -> cumulative knots in [-5,5]
        float mx = hj[0];
#pragma unroll
        for (int j = 1; j < 10; ++j) mx = fmaxf(mx, hj[j]);
        float e[10]; float s = 0.f;
#pragma unroll
        for (int j = 0; j < 10; ++j) { e[j] = expf((hj[j] - mx) * INV_SQRT_C); s += e[j]; }
        const float inv = 0.99f / s;        // (1 - min_bw*nb)/sum
        float c = 0.f; cw[0] = -5.f;
#pragma unroll
        for (int j = 0; j < 10; ++j) { c += 0.001f + e[j] * inv; cw[j + 1] = 10.f * c - 5.f; }
        cw[10] = 5.f;
    }
    {   // heights: hj[10:20]
        float mx = hj[10];
#pragma unroll
        for (int j = 11; j < 20; ++j) mx = fmaxf(mx, hj[j]);
        float e[10]; float s = 0.f;
#pragma unroll
        for (int j = 0; j < 10; ++j) { e[j] = expf((hj[10 + j] - mx) * INV_SQRT_C); s += e[j]; }
        const float inv = 0.99f / s;
        float c = 0.f; ch[0] = -5.f;
#pragma unroll
        for (int j = 0; j < 10; ++j) { c += 0.001f + e[j] * inv; ch[j + 1] = 10.f * c - 5.f; }
        ch[10] = 5.f;
    }
    // derivatives: pad ends -> softplus(log(expm1(1-min_d))) == 1-min_d -> 1.0
    dv[0] = 1.f; dv[10] = 1.f;
#pragma unroll
    for (int j = 0; j < 9; ++j) dv[j + 1] = 0.001f + softplusf(hj[20 + j]);

    const bool inside = (zb >= -5.f) && (zb <= 5.f);
    const float x = fminf(fmaxf(zb, -5.f), 5.f);

    int idx = -1;
#pragma unroll
    for (int i = 0; i < 11; ++i) {
        const float loc = (i == 10) ? (cw[10] + 1e-6f) : cw[i];
        idx += (x >= loc) ? 1 : 0;
    }
    idx = min(max(idx, 0), 9);

    float in_cw = 0.f, in_w = 1.f, in_ch = 0.f, in_h = 0.f, in_d = 1.f, in_d1 = 1.f;
#pragma unroll
    for (int i = 0; i < 10; ++i) {
        if (i == idx) {
            in_cw = cw[i]; in_w = cw[i + 1] - cw[i];
            in_ch = ch[i]; in_h = ch[i + 1] - ch[i];
            in_d  = dv[i]; in_d1 = dv[i + 1];
        }
    }
    const float delta = in_h / in_w;
    const float th  = (x - in_cw) / in_w;
    const float tmt = th * (1.f - th);
    const float num = in_h * (delta * th * th + in_d * tmt);
    const float den = delta + (in_d + in_d1 - 2.f * delta) * tmt;
    const float outv = in_ch + num / den;
    const float dnum = delta * delta *
        (in_d1 * th * th + 2.f * delta * tmt + in_d * (1.f - th) * (1.f - th));
    float lad = logf(dnum) - 2.f * logf(den);
    const float zb_new = inside ? outv : zb;
    lad = inside ? lad : 0.f;

    // z = concat(za, zb_new)*mask, then channel flip [:, ::-1]
    Z[((size_t)b * 2 + 0) * TT + t] = zb_new * m;
    Z[((size_t)b * 2 + 1) * TT + t] = za * m;
    blockAtomicAdd(&acc[b], lad * m, sred);
}

// --------- sigmoid/log midsection between post_flows and flows -------------
__global__ __launch_bounds__(256) void k_mid(
    float* __restrict__ Z, const float* __restrict__ Win,
    const float* __restrict__ mask, const float* __restrict__ em,
    const float* __restrict__ el, float* __restrict__ accLdq,
    float* __restrict__ accLd)
{
    __shared__ float sred[8];
    const int b = blockIdx.y;
    const int t = blockIdx.x * 256 + threadIdx.x;
    const float m  = mask[(size_t)b * TT + t];
    const float zu = Z[((size_t)b * 2 + 0) * TT + t];
    const float z1 = Z[((size_t)b * 2 + 1) * TT + t];
    const float u  = sigf(zu) * m;
    const float z0 = (Win[(size_t)b * TT + t] - u) * m;
    const float dldq = (logsigf(zu) + logsigf(-zu)) * m;
    const float z0l  = logf(fmaxf(z0, 1e-5f)) * m;
    const float el0 = el[0], el1 = el[1];
    Z[((size_t)b * 2 + 0) * TT + t] = (em[0] + expf(el0) * z0l) * m;
    Z[((size_t)b * 2 + 1) * TT + t] = (em[1] + expf(el1) * z1) * m;
    blockAtomicAdd(&accLdq[b], dldq, sred);
    blockAtomicAdd(&accLd[b], -z0l + (el0 + el1) * m, sred);
}

__global__ __launch_bounds__(256) void k_nll(
    const float* __restrict__ Z, const float* __restrict__ mask,
    float* __restrict__ accN)
{
    __shared__ float sred[8];
    const int b = blockIdx.y;
    const int t = blockIdx.x * 256 + threadIdx.x;
    const float m  = mask[(size_t)b * TT + t];
    const float z0 = Z[((size_t)b * 2 + 0) * TT + t];
    const float z1 = Z[((size_t)b * 2 + 1) * TT + t];
    blockAtomicAdd(&accN[b], 0.5f * (2.f * L2PI + z0 * z0 + z1 * z1) * m, sred);
}

__global__ __launch_bounds__(256) void k_final(
    const float* accN, const float* accLd, const float* accGq,
    const float* accLdq, float* __restrict__ out, int Bn)
{
    const int b = threadIdx.x;
    if (b < Bn) out[b] = (accN[b] - accLd[b]) + (accGq[b] - accLdq[b]);
}

// ------------------------------- host side ---------------------------------

extern "C" void kernel_launch(void* const* d_in, const int* in_sizes, int n_in,
                              void* d_out, int out_size, void* d_ws, size_t ws_size,
                              hipStream_t stream)
{
    const float* X  = (const float*)d_in[0];
    const float* M  = (const float*)d_in[1];
    const float* Wd = (const float*)d_in[2];
    const float* EQ = (const float*)d_in[3];
    const int Bn = in_sizes[0] / (CC * TT);

    // params flattened in JAX pytree order (dict keys sorted alphabetically)
    int cur = 4;
    auto nxt = [&]() { return (const float*)d_in[cur++]; };
    struct DdsL { const float *dw_b,*dw_w,*ln1_b,*ln1_g,*ln2_b,*ln2_g,*pw_b,*pw_w; };
    struct Flow { DdsL dds[3]; const float *in_b,*in_w,*proj_b,*proj_w; };
    auto readDds = [&](DdsL* L) {
        for (int i = 0; i < 3; ++i) {
            L[i].dw_b = nxt(); L[i].dw_w = nxt();
            L[i].ln1_b = nxt(); L[i].ln1_g = nxt();
            L[i].ln2_b = nxt(); L[i].ln2_g = nxt();
            L[i].pw_b = nxt(); L[i].pw_w = nxt();
        }
    };
    auto readFlow = [&](Flow& f) {
        readDds(f.dds);
        f.in_b = nxt(); f.in_w = nxt(); f.proj_b = nxt(); f.proj_w = nxt();
    };

    const float* conv1_b = nxt(); const float* conv1_w = nxt();
    const float* conv2_b = nxt(); const float* conv2_w = nxt();
    DdsL dds0[3]; readDds(dds0);
    const float* ewa_logs = nxt(); const float* ewa_m = nxt();
    Flow flows[4]; for (int i = 0; i < 4; ++i) readFlow(flows[i]);
    const float* pconv1_b = nxt(); const float* pconv1_w = nxt();
    const float* pconv2_b = nxt(); const float* pconv2_w = nxt();
    DdsL pdds[3]; readDds(pdds);
    const float* pewa_logs = nxt(); const float* pewa_m = nxt();
    Flow pflows[4]; for (int i = 0; i < 4; ++i) readFlow(pflows[i]);

    // workspace layout
    float* ws = (float*)d_ws;
    const size_t NA = (size_t)Bn * CC * TT;
    float* bufA  = ws;
    float* bufB  = ws + NA;
    float* bufC  = ws + 2 * NA;
    float* xg    = ws + 3 * NA;
    float* gpost = ws + 4 * NA;
    float* h29   = ws + 5 * NA;
    float* zbuf  = h29 + (size_t)Bn * 29 * TT;
    float* accLdq = zbuf + (size_t)Bn * 2 * TT;
    float* accLd  = accLdq + Bn;
    float* accGq  = accLd  + Bn;
    float* accN   = accGq  + Bn;

    const dim3 gElem((unsigned)((NA + 255) / 256));
    const dim3 gBT(TT / 256, Bn);
    const dim3 gTile(TT / 64, Bn);

    auto runDds = [&](float* Xb, const DdsL* L) {
        int dil = 1;
        for (int i = 0; i < 3; ++i) {
            k_dw_ln_gelu<<<gTile, 256, 0, stream>>>(
                Xb, M, L[i].dw_w, L[i].dw_b, L[i].ln1_g, L[i].ln1_b, bufB, dil);
            k_conv1x1_wmma<12, CC, false><<<gTile, 256, 0, stream>>>(
                bufB, L[i].pw_w, L[i].pw_b, nullptr, bufC);
            k_ln_gelu_add<<<gTile, 256, 0, stream>>>(
                bufC, L[i].ln2_g, L[i].ln2_b, M, Xb, (i == 2) ? 1 : 0);
            dil *= 3;
        }
    };

    k_zero<<<1, 256, 0, stream>>>(accLdq, 4 * Bn);

    // main x path: conv1 -> dds -> conv2  => xg (g for flows)
    k_conv1x1_wmma<12, CC, false><<<gTile, 256, 0, stream>>>(X, conv1_w, conv1_b, nullptr, bufA);
    runDds(bufA, dds0);
    k_conv1x1_wmma<12, CC, true><<<gTile, 256, 0, stream>>>(bufA, conv2_w, conv2_b, M, xg);

    // posterior path: expand(w) -> dds -> conv2; g_post = xg + h_w
    k_expand_add<<<gElem, 256, 0, stream>>>(Wd, TT, pconv1_w, pconv1_b, nullptr, bufA, (long long)NA);
    runDds(bufA, pdds);
    k_conv1x1_wmma<12, CC, true><<<gTile, 256, 0, stream>>>(bufA, pconv2_w, pconv2_b, M, bufB);
    k_add2<<<gElem, 256, 0, stream>>>(xg, bufB, gpost, (long long)NA);

    // z_q init (+ ewa-logs logdet term, gaussian logq term)
    k_zq_init<<<gBT, 256, 0, stream>>>(EQ, pewa_m, pewa_logs, M, zbuf, accLdq, accGq);

    // posterior flows
    for (int f = 0; f < 4; ++f) {
        k_expand_add<<<gElem, 256, 0, stream>>>(
            zbuf, 2 * TT, pflows[f].in_w, pflows[f].in_b, gpost, bufA, (long long)NA);
        runDds(bufA, pflows[f].dds);
        k_conv1x1_wmma<2, 29, true><<<gTile, 256, 0, stream>>>(
            bufA, pflows[f].proj_w, pflows[f].proj_b, M, h29);
        k_spline<<<gBT, 256, 0, stream>>>(zbuf, h29, M, accLdq);
    }

    // sigmoid / duration-log midsection, ewa re-scale
    k_mid<<<gBT, 256, 0, stream>>>(zbuf, Wd, M, ewa_m, ewa_logs, accLdq, accLd);

    // prior flows
    for (int f = 0; f < 4; ++f) {
        k_expand_add<<<gElem, 256, 0, stream>>>(
            zbuf, 2 * TT, flows[f].in_w, flows[f].in_b, xg, bufA, (long long)NA);
        runDds(bufA, flows[f].dds);
        k_conv1x1_wmma<2, 29, true><<<gTile, 256, 0, stream>>>(
            bufA, flows[f].proj_w, flows[f].proj_b, M, h29);
        k_spline<<<gBT, 256, 0, stream>>>(zbuf, h29, M, accLd);
    }

    // final gaussian NLL + combine:  out[b] = (nllG - logdet) + (gq - logdet_q)
    k_nll<<<gBT, 256, 0, stream>>>(zbuf, M, accN);
    k_final<<<1, 256, 0, stream>>>(accN, accLd, accGq, accLdq, (float*)d_out, Bn);
}